// GraphSAGE_54640573940274
// MI455X (gfx1250) — compile-verified
//
#include <hip/hip_runtime.h>

#define DFEAT 64

typedef __attribute__((ext_vector_type(2))) float v2f;
typedef __attribute__((ext_vector_type(8))) float v8f;

// ---------------------------------------------------------------- utilities
__global__ void sage_zero_f32(float* __restrict__ p, long n) {
  long i = (long)blockIdx.x * blockDim.x + threadIdx.x;
  long stride = (long)gridDim.x * blockDim.x;
  for (; i < n; i += stride) p[i] = 0.0f;
}

__global__ void sage_degree(const int* __restrict__ dst, int E,
                            float* __restrict__ deg) {
  int i = blockIdx.x * blockDim.x + threadIdx.x;
  if (i < E) atomicAdd(&deg[dst[i]], 1.0f);
}

// Scatter h[src[e]] into summed[dst[e]]; 64 consecutive lanes share one edge
// so feature loads and atomics are fully coalesced (edge ids are wave-uniform).
__global__ void sage_scatter(const float* __restrict__ h,
                             const int* __restrict__ src,
                             const int* __restrict__ dst,
                             long total, float* __restrict__ summed) {
  long i = (long)blockIdx.x * blockDim.x + threadIdx.x;
  long stride = (long)gridDim.x * blockDim.x;
  for (; i < total; i += stride) {
    long e = i >> 6;
    int d = (int)(i & 63);
    int s = src[e];
    int t = dst[e];
    atomicAdd(&summed[(long)t * DFEAT + d], h[(long)s * DFEAT + d]);
  }
}

// ------------------------------------------------- fused SAGE layer (WMMA)
// out[i] = act( h[i] @ W_self + (summed[i]/max(deg[i],1)) @ W_neigh + b )
// One block = 128 threads = 4 waves; block owns 16 node rows; wave w owns
// output columns [16w, 16w+16). K-loop: 16 steps of V_WMMA_F32_16X16X4_F32
// per weight matrix, accumulator chained across both matrices.
__global__ void __launch_bounds__(128)
sage_layer_wmma(const float* __restrict__ h,
                const float* __restrict__ summed,
                const float* __restrict__ deg,
                const float* __restrict__ Wself,
                const float* __restrict__ Wneigh,
                const float* __restrict__ bias,
                float* __restrict__ out,
                int n_nodes, int relu) {
  __shared__ float sWs[DFEAT * DFEAT];
  __shared__ float sWn[DFEAT * DFEAT];

  // cooperative stage of both 16KB weight matrices into LDS (float4 moves)
  for (int i = threadIdx.x; i < DFEAT * DFEAT / 4; i += blockDim.x) {
    ((float4*)sWs)[i] = ((const float4*)Wself)[i];
    ((float4*)sWn)[i] = ((const float4*)Wneigh)[i];
  }
  __syncthreads();

  const int lane  = threadIdx.x & 31;
  const int wave  = threadIdx.x >> 5;      // N-tile index 0..3
  const int m     = lane & 15;             // row within tile for A / col for B
  const int khalf = lane >> 4;             // 0: K pair (0,1), 1: K pair (2,3)
  const int row0  = blockIdx.x * 16;
  int row = row0 + m;
  if (row >= n_nodes) row = n_nodes - 1;   // uniform-safe clamp (N%16==0)
  const int n = wave * 16 + m;             // global output column

  // init accumulator with bias (depends only on column n)
  v8f c;
  const float bv = bias[n];
#pragma unroll
  for (int v = 0; v < 8; ++v) c[v] = bv;

  // ---- self term: h @ W_self
  const float* arow = h + (long)row * DFEAT;
#pragma unroll
  for (int kk = 0; kk < 16; ++kk) {
    const int kb = kk * 4 + khalf * 2;
    v2f a; a.x = arow[kb];           a.y = arow[kb + 1];
    v2f b; b.x = sWs[kb * DFEAT + n]; b.y = sWs[(kb + 1) * DFEAT + n];
    c = __builtin_amdgcn_wmma_f32_16x16x4_f32(false, a, false, b,
                                              (short)0, c, false, false);
  }

  // ---- neighbor term: (summed / max(deg,1)) @ W_neigh
  const float inv = 1.0f / fmaxf(deg[row], 1.0f);
  const float* srow = summed + (long)row * DFEAT;
#pragma unroll
  for (int kk = 0; kk < 16; ++kk) {
    const int kb = kk * 4 + khalf * 2;
    v2f a; a.x = srow[kb] * inv;      a.y = srow[kb + 1] * inv;
    v2f b; b.x = sWn[kb * DFEAT + n]; b.y = sWn[(kb + 1) * DFEAT + n];
    c = __builtin_amdgcn_wmma_f32_16x16x4_f32(false, a, false, b,
                                              (short)0, c, false, false);
  }

  // ---- store: C/D layout -> VGPR v holds M = v + 8*khalf, N = lane&15
#pragma unroll
  for (int v = 0; v < 8; ++v) {
    const int rm = row0 + v + khalf * 8;
    float val = c[v];
    if (relu) val = fmaxf(val, 0.0f);
    if (rm < n_nodes) out[(long)rm * DFEAT + n] = val;
  }
}

// ---------------------------------------------------------------- driver
static inline size_t align256(size_t x) { return (x + 255) & ~(size_t)255; }

extern "C" void kernel_launch(void* const* d_in, const int* in_sizes, int n_in,
                              void* d_out, int out_size, void* d_ws, size_t ws_size,
                              hipStream_t stream) {
  const float* x    = (const float*)d_in[0];
  const int*   src  = (const int*)d_in[1];
  const int*   dst  = (const int*)d_in[2];
  const float* W1s  = (const float*)d_in[3];
  const float* W1n  = (const float*)d_in[4];
  const float* b1   = (const float*)d_in[5];
  const float* W2s  = (const float*)d_in[6];
  const float* W2n  = (const float*)d_in[7];
  const float* b2   = (const float*)d_in[8];
  float* out = (float*)d_out;

  const int N = in_sizes[0] / DFEAT;
  const int E = in_sizes[1];

  // workspace: deg[N] | summed[N*D] | h1[N*D]
  char* ws = (char*)d_ws;
  float* deg    = (float*)ws;
  float* summed = (float*)(ws + align256((size_t)N * 4));
  float* h1     = (float*)(ws + align256((size_t)N * 4) +
                           align256((size_t)N * DFEAT * 4));

  const long nd = (long)N * DFEAT;
  const long ed = (long)E * DFEAT;

  const int ZB = 256;
  const int zeroBlocksND = (int)((nd + ZB - 1) / ZB);
  const int zeroBlocksN  = (N + ZB - 1) / ZB;
  const int scatBlocks   = (int)((ed + ZB - 1) / ZB);
  const int gemmBlocks   = (N + 15) / 16;

  // degree (shared by both layers)
  sage_zero_f32<<<zeroBlocksN, ZB, 0, stream>>>(deg, N);
  sage_degree<<<(E + ZB - 1) / ZB, ZB, 0, stream>>>(dst, E, deg);

  // ---- layer 1
  sage_zero_f32<<<zeroBlocksND, ZB, 0, stream>>>(summed, nd);
  sage_scatter<<<scatBlocks, ZB, 0, stream>>>(x, src, dst, ed, summed);
  sage_layer_wmma<<<gemmBlocks, 128, 0, stream>>>(x, summed, deg,
                                                  W1s, W1n, b1, h1,
                                                  N, /*relu=*/1);

  // ---- layer 2
  sage_zero_f32<<<zeroBlocksND, ZB, 0, stream>>>(summed, nd);
  sage_scatter<<<scatBlocks, ZB, 0, stream>>>(h1, src, dst, ed, summed);
  sage_layer_wmma<<<gemmBlocks, 128, 0, stream>>>(h1, summed, deg,
                                                  W2s, W2n, b2, out,
                                                  N, /*relu=*/0);
}